// cropnet_73306501808672
// MI455X (gfx1250) — compile-verified
//
#include <hip/hip_runtime.h>

// MI455X / gfx1250 crop+bilinear-resize kernel.
// Bilinear resize expressed as Out = R * Img * R^T and executed with
// V_WMMA_F32_16X16X4_F32 (wave32 WMMA). One workgroup per image,
// one wave per channel-tensor (3 color + 3 depth = 6 waves = 192 threads).

typedef __attribute__((ext_vector_type(2))) float v2f;
typedef __attribute__((ext_vector_type(8))) float v8f;

namespace {
constexpr int kH    = 224;
constexpr int kW    = 224;
constexpr int kPad  = 40;
constexpr int kOut  = 64;
constexpr int kTile = 32;   // padded crop side (w <= 26 < 32)
}

__global__ __launch_bounds__(192) void cropnet_wmma_kernel(
    const float* __restrict__ position,   // [B,3]
    const float* __restrict__ color_img,  // [B,3,224,224]
    const float* __restrict__ depth_img,  // [B,3,224,224]
    float* __restrict__ out_color,        // [B,3,64,64]
    float* __restrict__ out_depth,        // [B,3,64,64]
    float* __restrict__ out_ones)         // [B,1]
{
    __shared__ float sImg[6][kTile][kTile + 1];   // zero-padded crop tiles
    __shared__ float sU[6][kTile][kOut + 1];      // U = Img * R^T
    __shared__ float sF[kOut];
    __shared__ int   sI0[kOut];
    __shared__ int   sI1[kOut];

    const int b    = blockIdx.x;
    const int tid  = threadIdx.x;
    const int wave = tid >> 5;      // 0..5 -> channel-tensor
    const int lane = tid & 31;

    // ---- crop window parameters (uniform per block, recomputed per thread) ----
    const float px = position[3 * b + 0];
    const float py = position[3 * b + 1];
    const float pz = position[3 * b + 2];
    const float den = 1.73f * pz;
    const float pex = 1.5f * 224.0f * px / den;
    const float pey = 1.5f * 224.0f * py / den;
    const float ped = 1.5f * 224.0f * 0.1f / den;
    const int m      = (int)truncf(pex + 0.5f) + 112;
    const int n      = (int)truncf(pey + 0.5f) + 112;
    const int d_half = (int)truncf(ped * 0.5f) + 1;
    const int sx = kPad + m - d_half;
    const int sy = kPad + n - d_half;
    const int w  = 2 * d_half;            // 14..26

    // ---- bilinear LUT (same for x and y; square crop) ----
    if (tid < kOut) {
        float src = fmaxf((tid + 0.5f) * ((float)w / (float)kOut) - 0.5f, 0.0f);
        int i0 = (int)src;                 // src >= 0
        sF[tid]  = src - (float)i0;
        sI0[tid] = i0;
        sI1[tid] = min(i0 + 1, w - 1);
    }

    // ---- stage zero-padded 32x32 crop tiles (3 color + 3 depth) into LDS ----
    for (int e = tid; e < 6 * kTile * kTile; e += 192) {
        int c = e >> 10;                   // /1024
        int r = (e >> 5) & 31;
        int x = e & 31;
        float v = 0.0f;
        int oy = sy + r - kPad;
        int ox = sx + x - kPad;
        if (r < w && x < w && oy >= 0 && oy < kH && ox >= 0 && ox < kW) {
            const float* src = (c < 3) ? color_img : depth_img;
            v = src[(((size_t)b * 3 + (c % 3)) * kH + oy) * kW + ox];
        }
        sImg[c][r][x] = v;
    }
    __syncthreads();

    // WMMA fragment lane decomposition (32-bit 16x16x4 layouts, wave32):
    //   A: row = m0 + (lane&15), K pair = {khalf, khalf+1}
    //   B: col = n0 + (lane&15), K pair = {khalf, khalf+1}
    //   D: vgpr r -> row m0 + r + 8*(lane>=16), col n0 + (lane&15)
    const int c     = wave;
    const int nlo   = lane & 15;
    const int khalf = (lane >> 4) << 1;   // 0 or 2
    const int radd  = (lane >> 4) << 3;   // 0 or 8

    // ---- stage 1: U(32x64) = Img(32x32) * R^T(32x64); B synthesized from LUT ----
    for (int mt = 0; mt < 2; ++mt) {
        const int m0 = mt * 16;
        for (int nt = 0; nt < 4; ++nt) {
            const int n0   = nt * 16;
            const int jcol = n0 + nlo;       // output x index this lane produces
            const float f  = sF[jcol];
            const int i0   = sI0[jcol];
            const int i1   = sI1[jcol];
            v8f acc = {0.f, 0.f, 0.f, 0.f, 0.f, 0.f, 0.f, 0.f};
            #pragma unroll
            for (int kk = 0; kk < 8; ++kk) {
                const int kb = kk * 4 + khalf;
                v2f a, bf;
                a.x  = sImg[c][m0 + nlo][kb];
                a.y  = sImg[c][m0 + nlo][kb + 1];
                bf.x = (kb     == i0 ? 1.0f - f : 0.0f) + (kb     == i1 ? f : 0.0f);
                bf.y = (kb + 1 == i0 ? 1.0f - f : 0.0f) + (kb + 1 == i1 ? f : 0.0f);
                acc = __builtin_amdgcn_wmma_f32_16x16x4_f32(
                          false, a, false, bf, (short)0, acc, false, false);
            }
            #pragma unroll
            for (int r = 0; r < 8; ++r)
                sU[c][m0 + r + radd][n0 + nlo] = acc[r];
        }
    }
    // sU is produced and consumed by the same wave; LDS ops are in-order per wave.

    // ---- stage 2: Out(64x64) = R(64x32) * U(32x64); A synthesized from LUT ----
    float* outBase = ((c < 3) ? out_color : out_depth)
                     + (((size_t)b * 3 + (c % 3)) * kOut * kOut);
    for (int mt = 0; mt < 4; ++mt) {
        const int m0   = mt * 16;
        const int jrow = m0 + nlo;           // output y index this lane supplies
        const float f  = sF[jrow];
        const int i0   = sI0[jrow];
        const int i1   = sI1[jrow];
        for (int nt = 0; nt < 4; ++nt) {
            const int n0 = nt * 16;
            v8f acc = {0.f, 0.f, 0.f, 0.f, 0.f, 0.f, 0.f, 0.f};
            #pragma unroll
            for (int kk = 0; kk < 8; ++kk) {
                const int kb = kk * 4 + khalf;
                v2f a, bf;
                a.x  = (kb     == i0 ? 1.0f - f : 0.0f) + (kb     == i1 ? f : 0.0f);
                a.y  = (kb + 1 == i0 ? 1.0f - f : 0.0f) + (kb + 1 == i1 ? f : 0.0f);
                bf.x = sU[c][kb][n0 + nlo];
                bf.y = sU[c][kb + 1][n0 + nlo];
                acc = __builtin_amdgcn_wmma_f32_16x16x4_f32(
                          false, a, false, bf, (short)0, acc, false, false);
            }
            #pragma unroll
            for (int r = 0; r < 8; ++r)
                outBase[(m0 + r + radd) * kOut + n0 + nlo] = acc[r];
        }
    }

    if (tid == 0) out_ones[b] = 1.0f;
}

extern "C" void kernel_launch(void* const* d_in, const int* in_sizes, int n_in,
                              void* d_out, int out_size, void* d_ws, size_t ws_size,
                              hipStream_t stream) {
    (void)n_in; (void)out_size; (void)d_ws; (void)ws_size;
    const float* position = (const float*)d_in[0];
    const float* color    = (const float*)d_in[1];
    const float* depth    = (const float*)d_in[2];
    const int B = in_sizes[0] / 3;                      // 2048

    float* out       = (float*)d_out;
    float* out_color = out;
    float* out_depth = out + (size_t)B * 3 * kOut * kOut;
    float* out_ones  = out + (size_t)B * 3 * kOut * kOut * 2;

    cropnet_wmma_kernel<<<B, 192, 0, stream>>>(position, color, depth,
                                               out_color, out_depth, out_ones);
}